// SAMMultiLevelAssignLoss_25220047962854
// MI455X (gfx1250) — compile-verified
//
#include <hip/hip_runtime.h>
#include <hip/hip_bf16.h>
#include <stdint.h>

// Problem geometry (fixed by the reference).
#define B_      32
#define K_      4
#define HW_     262144      // 512*512
#define TILE_   1024        // pixels staged per tile (256 thr * 4 px)
#define ITER_   8           // tiles per block
#define CHUNKS_ 32          // CHUNKS_*ITER_*TILE_ == HW_
#define SLOTS_  4           // LDS pipeline depth (3 tiles in flight + 1 consuming)

#define ALPHA_   0.8f
#define SMOOTH_  1e-4f
#define FOCAL_W_ 20.0f

// ws layout (floats): [(b*K+k)*5 + {0:focal,1:psum,2:inter,3:msum,4:mi}] then tsum[b]
#define WS_TSUM_ (B_ * K_ * 5)
#define WS_N_    (WS_TSUM_ + B_)

// ---- CDNA5 async global->LDS path (guarded; falls back to float4 streaming) ----
#if __has_builtin(__builtin_amdgcn_global_load_async_to_lds_b128)
#define USE_ASYNC 1
#else
#define USE_ASYNC 0
#endif

#if __has_builtin(__builtin_amdgcn_s_wait_asynccnt)
#define WAIT_ASYNC(n) __builtin_amdgcn_s_wait_asynccnt(n)
#else
#define WAIT_ASYNC(n) asm volatile("s_wait_asynccnt " #n ::: "memory")
#endif

// void __builtin_amdgcn_global_load_async_to_lds_b128(AS1 int4* src, AS3 int4* dst,
//                                                     int offset, int cpol)
typedef int v4i_ __attribute__((vector_size(16)));
typedef __attribute__((address_space(1))) v4i_* gv4_ptr;
typedef __attribute__((address_space(3))) v4i_* lv4_ptr;

#define ASYNC_CP16(dst_lds, src_g)                                   \
  __builtin_amdgcn_global_load_async_to_lds_b128(                    \
      (gv4_ptr)(src_g), (lv4_ptr)(dst_lds), 0, 0)

// Per-pixel loss contributions, built on raw v_exp/v_log/v_rcp.
// With e = exp(-|x|) in (0,1], inv = 1/(1+e):
//   sigmoid(x)   = x>=0 ? inv    : e*inv
//   1-sigmoid(x) = x>=0 ? e*inv  : inv        (exact complement, no cancellation)
//   pt  = exp(-bce) = t ? sig : 1-sig         (t in {0,1})
//   bce = -log(pt);  focal integrand = ALPHA*(1-pt)^2*bce
__device__ __forceinline__ void px_accum(float x, float t,
                                         float& f, float& p, float& i,
                                         float& m, float& mi) {
  float ax   = fminf(fabsf(x), 80.0f);          // clamp: keep e > 0
  float e    = __expf(-ax);                      // v_exp_f32 path
  float inv  = __builtin_amdgcn_rcpf(1.0f + e);  // raw v_rcp_f32
  float ei   = e * inv;
  bool  pos  = (x >= 0.0f);                      // MASK_THR == 0
  float sig  = pos ? inv : ei;                   // sigmoid(x)
  float oneg = pos ? ei : inv;                   // 1 - sigmoid(x)
  bool  tp   = (t > 0.5f);
  float pt   = tp ? sig : oneg;                  // exp(-bce)
  float om   = tp ? oneg : sig;                  // 1 - pt (exact)
  float bce  = -__logf(pt);
  f  += ALPHA_ * om * om * bce;                  // GAMMA == 2
  p  += sig;
  i  += sig * t;
  float mm = pos ? 1.0f : 0.0f;
  m  += mm;
  mi += mm * t;
}

__global__ void init_ws(float* __restrict__ ws) {
  int i = blockIdx.x * blockDim.x + threadIdx.x;
  if (i < WS_N_) ws[i] = 0.0f;
}

__global__ __launch_bounds__(256) void partial_kernel(
    const float* __restrict__ pred, const float* __restrict__ tgt,
    float* __restrict__ ws) {
  const int tid   = threadIdx.x;
  const int b     = blockIdx.x / CHUNKS_;
  const int chunk = blockIdx.x % CHUNKS_;
  const int base0 = chunk * (ITER_ * TILE_);

  const float* tb = tgt  + (size_t)b * HW_;
  const float* xb = pred + (size_t)b * (K_ * HW_);

  float aF[K_]  = {0.f, 0.f, 0.f, 0.f};
  float aP[K_]  = {0.f, 0.f, 0.f, 0.f};
  float aI[K_]  = {0.f, 0.f, 0.f, 0.f};
  float aM[K_]  = {0.f, 0.f, 0.f, 0.f};
  float aMI[K_] = {0.f, 0.f, 0.f, 0.f};
  float aT = 0.f;

#if USE_ASYNC
  __shared__ float sT[SLOTS_][TILE_];
  __shared__ float sX[SLOTS_][K_][TILE_];

  // Stage one tile (1 target b128 + 4 pred b128 per thread) into LDS slot s.
  auto stage = [&](int s, int o) {
    const float* gt = tb + o + tid * 4;
    ASYNC_CP16(&sT[s][tid * 4], gt);
#pragma unroll
    for (int k = 0; k < K_; ++k) {
      const float* gx = xb + (size_t)k * HW_ + o + tid * 4;
      ASYNC_CP16(&sX[s][k][tid * 4], gx);
    }
  };

  stage(0, base0);                         // 3-deep prologue: 15 b128s in flight
  stage(1, base0 + 1 * TILE_);
  stage(2, base0 + 2 * TILE_);
  for (int it = 0; it < ITER_; ++it) {
    const int ahead = it + (SLOTS_ - 1);
    if (ahead < ITER_) {
      stage(ahead & (SLOTS_ - 1), base0 + ahead * TILE_);
      WAIT_ASYNC(15);                      // 3 tiles (15 loads) may stay in flight
    } else if (ahead == ITER_) {
      WAIT_ASYNC(10);
    } else if (ahead == ITER_ + 1) {
      WAIT_ASYNC(5);
    } else {
      WAIT_ASYNC(0);
    }
    asm volatile("" ::: "memory");         // order LDS reads after the wait
    const int s = it & (SLOTS_ - 1);
    float4 t4 = *(const float4*)&sT[s][tid * 4];
#pragma unroll
    for (int k = 0; k < K_; ++k) {
      float4 x4 = *(const float4*)&sX[s][k][tid * 4];
      px_accum(x4.x, t4.x, aF[k], aP[k], aI[k], aM[k], aMI[k]);
      px_accum(x4.y, t4.y, aF[k], aP[k], aI[k], aM[k], aMI[k]);
      px_accum(x4.z, t4.z, aF[k], aP[k], aI[k], aM[k], aMI[k]);
      px_accum(x4.w, t4.w, aF[k], aP[k], aI[k], aM[k], aMI[k]);
    }
    aT += t4.x + t4.y + t4.z + t4.w;
  }
#else
  for (int it = 0; it < ITER_; ++it) {
    const int o = base0 + it * TILE_ + tid * 4;
    float4 t4 = *(const float4*)(tb + o);
#pragma unroll
    for (int k = 0; k < K_; ++k) {
      float4 x4 = *(const float4*)(xb + (size_t)k * HW_ + o);
      px_accum(x4.x, t4.x, aF[k], aP[k], aI[k], aM[k], aMI[k]);
      px_accum(x4.y, t4.y, aF[k], aP[k], aI[k], aM[k], aMI[k]);
      px_accum(x4.z, t4.z, aF[k], aP[k], aI[k], aM[k], aMI[k]);
      px_accum(x4.w, t4.w, aF[k], aP[k], aI[k], aM[k], aMI[k]);
    }
    aT += t4.x + t4.y + t4.z + t4.w;
  }
#endif

  // --- wave32 shuffle reduction -> LDS -> 21 global f32 atomics per block ---
  float vals[21];
#pragma unroll
  for (int k = 0; k < K_; ++k) {
    vals[k * 5 + 0] = aF[k];
    vals[k * 5 + 1] = aP[k];
    vals[k * 5 + 2] = aI[k];
    vals[k * 5 + 3] = aM[k];
    vals[k * 5 + 4] = aMI[k];
  }
  vals[20] = aT;

  __shared__ float red[21];
  if (tid < 21) red[tid] = 0.0f;
  __syncthreads();
#pragma unroll
  for (int q = 0; q < 21; ++q) {
    float v = vals[q];
#pragma unroll
    for (int off = 16; off > 0; off >>= 1) v += __shfl_xor(v, off, 32);
    if ((tid & 31) == 0) atomicAdd(&red[q], v);
  }
  __syncthreads();
  if (tid < 20) {
    atomicAdd(&ws[(b * K_ + tid / 5) * 5 + (tid % 5)], red[tid]);
  } else if (tid == 20) {
    atomicAdd(&ws[WS_TSUM_ + b], red[20]);
  }
}

__global__ void finalize_kernel(const float* __restrict__ ws,
                                const float* __restrict__ pious,
                                float* __restrict__ out) {
  if (threadIdx.x != 0 || blockIdx.x != 0) return;
  const float LO[4] = {0.04f, 0.0f, 0.01f, 0.16f};
  const float HI[4] = {0.64f, 0.04f, 0.25f, 1.0f};
  float aggF = 0.f, aggD = 0.f, aggI = 0.f, nval = 0.f;
  for (int b = 0; b < B_; ++b) {
    float ts    = ws[WS_TSUM_ + b];
    float ratio = ts / (float)HW_;
    float cf = 0.f, cd = 0.f, ci = 0.f, cnt = 0.f;
    for (int k = 0; k < K_; ++k) {
      const float* s = &ws[(b * K_ + k) * 5];
      float fm   = s[0] / (float)HW_;                             // focal mean
      float dice = 1.0f - (2.0f * s[2] + SMOOTH_) / (s[1] + ts + SMOOTH_);
      float iou_gt = (s[4] + SMOOTH_) / (s[3] + ts - s[4] + SMOOTH_);
      float d    = pious[b * K_ + k] - iou_gt;
      bool  v    = (ratio > LO[k]) && (ratio < HI[k]);
      if (v) { cf += fm; cd += dice; ci += d * d; cnt += 1.0f; }
    }
    if (cnt > 0.0f) {
      aggF += cf / cnt; aggD += cd / cnt; aggI += ci / cnt; nval += 1.0f;
    }
  }
  float inv = (nval > 0.0f) ? (1.0f / nval) : 1.0f;
  out[0] = FOCAL_W_ * aggF * inv;
  out[1] = aggD * inv;
  out[2] = aggI * inv;
}

extern "C" void kernel_launch(void* const* d_in, const int* in_sizes, int n_in,
                              void* d_out, int out_size, void* d_ws, size_t ws_size,
                              hipStream_t stream) {
  (void)in_sizes; (void)n_in; (void)out_size; (void)ws_size;
  const float* pred  = (const float*)d_in[0];   // pred_masks (32,4,512,512)
  const float* pious = (const float*)d_in[1];   // pred_ious  (32,4)
  const float* tgt   = (const float*)d_in[2];   // targets    (32,1,512,512)
  float* out = (float*)d_out;
  float* ws  = (float*)d_ws;

  init_ws<<<(WS_N_ + 255) / 256, 256, 0, stream>>>(ws);
  partial_kernel<<<B_ * CHUNKS_, 256, 0, stream>>>(pred, tgt, ws);
  finalize_kernel<<<1, 64, 0, stream>>>(ws, pious, out);
}